// GaussianGCN_35029753266633
// MI455X (gfx1250) — compile-verified
//
#include <hip/hip_runtime.h>
#include <hip/hip_bf16.h>
#include <math.h>

// GaussianGCN fused pipeline for MI455X (gfx1250, wave32).
// All matmuls run on V_WMMA_F32_16X16X4_F32 (full fp32 precision; d2 is a
// cancellation-sensitive expression feeding exp()). Everything is L2-resident.

#define BB 2
#define CC 256
#define NN 4096
#define NEG_INV_2PI (-0.15915494309189535f)

typedef float v2f __attribute__((ext_vector_type(2)));
typedef float v8f __attribute__((ext_vector_type(8)));

#define WMMA_F32(a, b, c) \
  __builtin_amdgcn_wmma_f32_16x16x4_f32(false, (a), false, (b), (short)0, (c), false, false)

// ---------------------------------------------------------------- zero init
__global__ void zero_kernel(float* __restrict__ p, long count) {
  long i = (long)blockIdx.x * blockDim.x + threadIdx.x;
  if (i < count) p[i] = 0.0f;
}

// ------------------------------------------------- tiled transpose (R x Cc -> Cc x R)
__global__ void transpose_kernel(const float* __restrict__ src, float* __restrict__ dst,
                                 int R, int Cc) {
  __shared__ float tile[32][33];
  const float* s = src + (size_t)blockIdx.z * R * Cc;
  float* d = dst + (size_t)blockIdx.z * R * Cc;
  int x0 = blockIdx.x * 32, y0 = blockIdx.y * 32;
  for (int i = threadIdx.y; i < 32; i += blockDim.y) {
    int r = y0 + i, c = x0 + threadIdx.x;
    tile[i][threadIdx.x] = (r < R && c < Cc) ? s[(size_t)r * Cc + c] : 0.0f;
  }
  __syncthreads();
  for (int i = threadIdx.y; i < 32; i += blockDim.y) {
    int r = x0 + i, c = y0 + threadIdx.x;      // dst is Cc x R
    if (r < Cc && c < R) d[(size_t)r * R + c] = tile[threadIdx.x][i];
  }
}

// ---------------------------------------------------------------- row norms
__global__ void sq_kernel(const float* __restrict__ x, float* __restrict__ sq) {
  int i = blockIdx.x * blockDim.x + threadIdx.x;   // over B*N
  if (i >= BB * NN) return;
  int b = i >> 12, n = i & (NN - 1);
  const float* Xb = x + (size_t)b * CC * NN;
  float s = 0.0f;
  for (int c = 0; c < CC; ++c) { float v = Xb[(size_t)c * NN + n]; s += v * v; }
  sq[i] = s;
}

// --------------------------------------- pass 1: column sums of AV (fused gram+exp)
// One wave = 64 rows x 16 cols of AV; B-fragments shared across the 4 row tiles.
__global__ __launch_bounds__(256) void colsum_kernel(const float* __restrict__ x,
                                                     const float* __restrict__ sq,
                                                     float* __restrict__ colsum) {
  const int lane = threadIdx.x & 31;
  const int wid  = threadIdx.x >> 5;
  int unit = blockIdx.x * 8 + wid;            // B * 256 col-tiles * 64 row-blocks
  int rb = unit & 63;
  int ct = (unit >> 6) & 255;
  int b  = unit >> 14;
  const float* Xb  = x  + (size_t)b * CC * NN;
  const float* sqb = sq + b * NN;
  const int lm = lane & 15;
  const int kh = (lane >> 4) << 1;
  const int hi8 = (lane >> 4) << 3;
  const int cbase = ct * 16;
  const int rbase0 = rb * 64;
  float sqc = sqb[cbase + lm];

  v8f acc[4];
  #pragma unroll
  for (int t = 0; t < 4; ++t) acc[t] = (v8f){0,0,0,0,0,0,0,0};

  for (int s = 0; s < 64; ++s) {
    const float* pk = Xb + (size_t)((s << 2) + kh) * NN;
    v2f bf = { pk[cbase + lm], pk[NN + cbase + lm] };
    #pragma unroll
    for (int t = 0; t < 4; ++t) {
      int r = rbase0 + t * 16 + lm;
      v2f af = { pk[r], pk[NN + r] };
      acc[t] = WMMA_F32(af, bf, acc[t]);
    }
  }
  float csum = 0.0f;
  #pragma unroll
  for (int t = 0; t < 4; ++t) {
    int r0 = rbase0 + t * 16 + hi8;
    #pragma unroll
    for (int g = 0; g < 8; ++g) {
      float d2 = sqb[r0 + g] + sqc - 2.0f * acc[t][g];
      csum += __expf(d2 * NEG_INV_2PI);
    }
  }
  csum += __shfl_xor(csum, 16, 32);
  if (lane < 16) atomicAdd(&colsum[b * NN + cbase + lane], csum);
}

__global__ void deg_kernel(const float* __restrict__ colsum, float* __restrict__ deg) {
  int i = blockIdx.x * blockDim.x + threadIdx.x;
  if (i < BB * NN) deg[i] = rsqrtf(1.0f + colsum[i]);
}

// ----------------- pass 2: agg = D(A+I)D * node_k, fused gram -> exp -> LDS -> GEMM2
// One wave owns a 16-row n-tile across ALL 256 channels (16 v8f accumulators),
// loops over a quarter of the m range; AV tile staged through LDS to convert
// accumulator (D) layout into A-fragment layout.
__global__ __launch_bounds__(256) void agg_kernel(const float* __restrict__ x,
                                                  const float* __restrict__ nodeT,
                                                  const float* __restrict__ sq,
                                                  const float* __restrict__ deg,
                                                  float* __restrict__ agg) {
  __shared__ float lds[8][16 * 17];
  const int lane = threadIdx.x & 31;
  const int wid  = threadIdx.x >> 5;
  int unit = blockIdx.x * 8 + wid;            // B * 256 n-tiles * 4 m-splits
  int ms = unit & 3;
  int nt = (unit >> 2) & 255;
  int b  = unit >> 10;
  const float* Xb   = x     + (size_t)b * CC * NN;
  const float* Tb   = nodeT + (size_t)b * NN * CC;
  const float* sqb  = sq  + b * NN;
  const float* degb = deg + b * NN;
  const int lm = lane & 15;
  const int kh  = (lane >> 4) << 1;
  const int hi8 = (lane >> 4) << 3;
  const int nbase = nt * 16;
  float* myl = lds[wid];

  float sqn[8], dgn[8];
  #pragma unroll
  for (int g = 0; g < 8; ++g) {
    sqn[g] = sqb[nbase + hi8 + g];
    dgn[g] = degb[nbase + hi8 + g];
  }
  v8f cacc[16];
  #pragma unroll
  for (int j = 0; j < 16; ++j) cacc[j] = (v8f){0,0,0,0,0,0,0,0};

  for (int mc = ms * 64; mc < ms * 64 + 64; ++mc) {
    int mbase = mc * 16;
    v8f g8 = (v8f){0,0,0,0,0,0,0,0};
    for (int s = 0; s < 64; ++s) {
      const float* pk = Xb + (size_t)((s << 2) + kh) * NN;
      v2f af = { pk[nbase + lm], pk[NN + nbase + lm] };
      v2f bf = { pk[mbase + lm], pk[NN + mbase + lm] };
      g8 = WMMA_F32(af, bf, g8);
    }
    float sqm = sqb[mbase + lm];
    float dgm = degb[mbase + lm];
    #pragma unroll
    for (int g = 0; g < 8; ++g) {
      float d2 = sqn[g] + sqm - 2.0f * g8[g];
      float av = __expf(d2 * NEG_INV_2PI);
      if (nbase + hi8 + g == mbase + lm) av += 1.0f;   // + identity
      av *= dgn[g] * dgm;
      myl[(hi8 + g) * 17 + lm] = av;                   // stage AVn tile [n][m]
    }
    // second GEMM: cacc[j] += AVtile(16x16) * node_k[mbase..][c-tile j]
    #pragma unroll
    for (int s2 = 0; s2 < 4; ++s2) {
      int k0 = (s2 << 2) + kh;
      v2f a2 = { myl[lm * 17 + k0], myl[lm * 17 + k0 + 1] };
      const float* pT = Tb + (size_t)(mbase + k0) * CC;
      #pragma unroll
      for (int j = 0; j < 16; ++j) {
        v2f b2 = { pT[j * 16 + lm], pT[CC + j * 16 + lm] };
        cacc[j] = WMMA_F32(a2, b2, cacc[j]);
      }
    }
  }
  #pragma unroll
  for (int j = 0; j < 16; ++j)
    #pragma unroll
    for (int g = 0; g < 8; ++g)
      atomicAdd(&agg[(size_t)(b * NN + nbase + hi8 + g) * CC + j * 16 + lm], cacc[j][g]);
}

// -------------------------- pass 3: AVW = agg * W^T + b  (+ BatchNorm statistics)
__global__ __launch_bounds__(256) void linear_bn_kernel(const float* __restrict__ agg,
                                                        const float* __restrict__ Wt,
                                                        const float* __restrict__ bias,
                                                        float* __restrict__ AVW,
                                                        float* __restrict__ chanSum,
                                                        float* __restrict__ chanSumSq) {
  const int lane = threadIdx.x & 31;
  const int wid  = threadIdx.x >> 5;
  int unit = blockIdx.x * 8 + wid;            // 512 row-tiles * 16 col-tiles
  int jt = unit & 15;
  int rt = unit >> 4;
  const int lm = lane & 15;
  const int kh  = (lane >> 4) << 1;
  const int hi8 = (lane >> 4) << 3;
  int rbase = rt * 16, cbase = jt * 16;

  v8f acc = (v8f){0,0,0,0,0,0,0,0};
  for (int s = 0; s < 64; ++s) {
    int k = (s << 2) + kh;
    const float* pa = agg + (size_t)(rbase + lm) * CC + k;
    v2f af = { pa[0], pa[1] };
    const float* pw = Wt + (size_t)k * CC;
    v2f bf = { pw[cbase + lm], pw[CC + cbase + lm] };
    acc = WMMA_F32(af, bf, acc);
  }
  float bv = bias[cbase + lm];
  float s1 = 0.0f, s2 = 0.0f;
  #pragma unroll
  for (int g = 0; g < 8; ++g) {
    float v = acc[g] + bv;
    AVW[(size_t)(rbase + hi8 + g) * CC + cbase + lm] = v;
    s1 += v;
    s2 += v * v;
  }
  s1 += __shfl_xor(s1, 16, 32);
  s2 += __shfl_xor(s2, 16, 32);
  if (lane < 16) {
    atomicAdd(&chanSum[cbase + lane], s1);
    atomicAdd(&chanSumSq[cbase + lane], s2);
  }
}

// -------------------------------- pass 4: BatchNorm apply + [b,n,c] -> [b,c,n]
__global__ void finalize_kernel(const float* __restrict__ AVW,
                                const float* __restrict__ chanSum,
                                const float* __restrict__ chanSumSq,
                                const float* __restrict__ gamma,
                                const float* __restrict__ beta,
                                float* __restrict__ out) {
  int o = blockIdx.x * blockDim.x + threadIdx.x;   // B*C*N = 2^21
  if (o >= BB * CC * NN) return;
  int n = o & (NN - 1);
  int c = (o >> 12) & (CC - 1);
  int b = o >> 20;
  const float invBN = 1.0f / (BB * NN);
  float mean = chanSum[c] * invBN;
  float var  = chanSumSq[c] * invBN - mean * mean;
  float inv  = rsqrtf(var + 1e-5f);
  float v = AVW[(size_t)(b * NN + n) * CC + c];
  out[o] = gamma[c] * (v - mean) * inv + beta[c];
}

// ------------------------------------------------------------------ launcher
extern "C" void kernel_launch(void* const* d_in, const int* in_sizes, int n_in,
                              void* d_out, int out_size, void* d_ws, size_t ws_size,
                              hipStream_t stream) {
  (void)in_sizes; (void)n_in; (void)out_size; (void)ws_size;
  const float* x     = (const float*)d_in[0];
  const float* W     = (const float*)d_in[1];
  const float* b_lin = (const float*)d_in[2];
  const float* gamma = (const float*)d_in[3];
  const float* beta  = (const float*)d_in[4];
  float* out = (float*)d_out;

  // workspace layout (floats)
  float* ws        = (float*)d_ws;
  float* sq        = ws;                        //  8192
  float* colsum    = sq + BB * NN;              //  8192   (zeroed)
  float* chanSum   = colsum + BB * NN;          //   256   (zeroed)
  float* chanSumSq = chanSum + CC;              //   256   (zeroed)
  float* agg       = chanSumSq + CC;            //  2M     (zeroed)
  float* deg       = agg + (size_t)BB * NN * CC;//  8192
  float* nodeT     = deg + BB * NN;             //  2M
  float* Wt        = nodeT + (size_t)BB * NN * CC; // 64K
  float* AVW       = nodeT;                     // alias: nodeT dead after agg pass

  // 1) zero accumulator regions (colsum, chanSum, chanSumSq, agg are contiguous)
  long zcount = (long)BB * NN + 2 * CC + (long)BB * NN * CC;
  zero_kernel<<<(int)((zcount + 255) / 256), 256, 0, stream>>>(colsum, zcount);

  // 2) transposes: x [C][N] -> nodeT [N][C] per batch; W [C][C] -> Wt
  transpose_kernel<<<dim3(NN / 32, CC / 32, BB), dim3(32, 8), 0, stream>>>(x, nodeT, CC, NN);
  transpose_kernel<<<dim3(CC / 32, CC / 32, 1), dim3(32, 8), 0, stream>>>(W, Wt, CC, CC);

  // 3) row norms
  sq_kernel<<<(BB * NN) / 256, 256, 0, stream>>>(x, sq);

  // 4) colsum of AV (fused gram+exp): B*256*64 waves, 8 waves/block
  colsum_kernel<<<(BB * 256 * 64) / 8, 256, 0, stream>>>(x, sq, colsum);

  // 5) deg = rsqrt(1 + colsum)
  deg_kernel<<<(BB * NN) / 256, 256, 0, stream>>>(colsum, deg);

  // 6) agg = D(A+I)D * node_k : B*256*4 waves, 8 waves/block
  agg_kernel<<<(BB * 256 * 4) / 8, 256, 0, stream>>>(x, nodeT, sq, deg, agg);

  // 7) linear + BN stats: 512*16 waves, 8 waves/block
  linear_bn_kernel<<<(512 * 16) / 8, 256, 0, stream>>>(agg, Wt, b_lin, AVW, chanSum, chanSumSq);

  // 8) BN apply + transpose to output layout
  finalize_kernel<<<(BB * CC * NN) / 256, 256, 0, stream>>>(AVW, chanSum, chanSumSq,
                                                            gamma, beta, out);
}